// Gibbs_34583076667752
// MI455X (gfx1250) — compile-verified
//
#include <hip/hip_runtime.h>
#include <math.h>

// ---------------------------------------------------------------------------
// LJ pairwise energy on MI455X (gfx1250), wave32.
//
// r2(i,j) = |xi|^2 + |xj|^2 - 2*dot(xi,xj).  dot computed for 16x16 tiles via
// V_WMMA_F32_16X16X4_F32 (D=3 padded to K=4 with zeros; exact in fp32).
// Epilogue per pair (branch-free): r2 = clip(r2, 1e-10);
// inv = rcp(r2) + 1 Newton step; sr6 = (sigma^2 * inv)^3;
// acc += weight * (sr6^2 - sr6), weight = (i<j && mask_i && mask_j) ? 4eps : 0.
// Output[b] = -sum(phi).
// ---------------------------------------------------------------------------

#define BATCH 8
#define NPART 2048
#define TILE 16
#define NT (NPART / TILE)                 // 128 tiles per dimension
#define NPAIR_TILES (NT * (NT + 1) / 2)   // 8256 upper-triangular tile pairs
#define WAVES_PER_BLOCK 8
#define NBLOCKS_X (NPAIR_TILES / WAVES_PER_BLOCK)  // 1032, exact
#define R2MIN 1e-10f

typedef __attribute__((ext_vector_type(2))) float v2f;
typedef __attribute__((ext_vector_type(8))) float v8f;

__global__ __launch_bounds__(256) void lj_tile_kernel(
    const float* __restrict__ x,            // [B, N, 3]
    const unsigned char* __restrict__ mask, // [B, N] bool (1 byte)
    const float* __restrict__ sigma_raw,    // [1]
    const float* __restrict__ eps_raw,      // [1]
    float* __restrict__ partial)            // [B, NBLOCKS_X]
{
  const int w    = threadIdx.x >> 5;      // wave id in block (0..7)
  const int lane = threadIdx.x & 31;
  const int b    = blockIdx.y;
  const int t    = blockIdx.x * WAVES_PER_BLOCK + w;  // < NPAIR_TILES always

  // Decode t -> (ti, tj), 0 <= ti <= tj < NT, row-major upper triangle.
  // Row ti starts at S(ti) = ti*NT - ti*(ti-1)/2.
  int ti = (int)((2.0f * NT + 1.0f -
                  sqrtf((2.0f * NT + 1.0f) * (2.0f * NT + 1.0f) - 8.0f * (float)t)) * 0.5f);
  if (ti < 0) ti = 0;
  if (ti > NT - 1) ti = NT - 1;
  while (ti > 0 && (ti * NT - (ti * (ti - 1)) / 2) > t) --ti;
  while (((ti + 1) * NT - ((ti + 1) * ti) / 2) <= t) ++ti;
  const int row_start = ti * NT - (ti * (ti - 1)) / 2;
  const int tj = ti + (t - row_start);

  const int i0 = ti * TILE;
  const int j0 = tj * TILE;

  const float* xb = x + (size_t)b * NPART * 3;
  const unsigned char* mb = mask + (size_t)b * NPART;

  __shared__ float s_sqi[WAVES_PER_BLOCK][TILE];
  __shared__ float s_sqj[WAVES_PER_BLOCK][TILE];
  __shared__ float s_mi [WAVES_PER_BLOCK][TILE];   // 1.0 / 0.0 weights
  __shared__ float s_mj [WAVES_PER_BLOCK][TILE];
  __shared__ float s_red[WAVES_PER_BLOCK];

  const int  r  = lane & 15;
  const bool hi = lane >= 16;

  // Stage 1: publish |x|^2 and mask weights for this wave's 16 rows / cols.
  if (!hi) {
    const float* p = xb + (size_t)(i0 + r) * 3;
    const float x0 = p[0], x1 = p[1], x2 = p[2];
    s_sqi[w][r] = x0 * x0 + x1 * x1 + x2 * x2;
    s_mi [w][r] = mb[i0 + r] ? 1.0f : 0.0f;
  } else {
    const float* p = xb + (size_t)(j0 + r) * 3;
    const float x0 = p[0], x1 = p[1], x2 = p[2];
    s_sqj[w][r] = x0 * x0 + x1 * x1 + x2 * x2;
    s_mj [w][r] = mb[j0 + r] ? 1.0f : 0.0f;
  }
  __syncthreads();

  // Stage 2: load WMMA fragments.
  // A (16x4 f32): lanes 0-15 hold (K0,K1) of row M=lane; lanes 16-31 hold
  // (K2,K3) of row M=lane-16. K3 is the zero pad. B (4x16) mirrors this
  // layout with columns across lanes, so the per-lane load pattern matches.
  v2f afrag, bfrag;
  {
    const float* pa = xb + (size_t)(i0 + r) * 3;
    const float* pb = xb + (size_t)(j0 + r) * 3;
    if (!hi) {
      afrag.x = pa[0]; afrag.y = pa[1];
      bfrag.x = pb[0]; bfrag.y = pb[1];
    } else {
      afrag.x = pa[2]; afrag.y = 0.0f;
      bfrag.x = pb[2]; bfrag.y = 0.0f;
    }
  }

  v8f c = {0.f, 0.f, 0.f, 0.f, 0.f, 0.f, 0.f, 0.f};
  // D = A x B + C ; EXEC is all-ones here (no divergence at this point).
  c = __builtin_amdgcn_wmma_f32_16x16x4_f32(
        /*neg_a=*/false, afrag, /*neg_b=*/false, bfrag,
        /*c_mod=*/(short)0, c, /*reuse_a=*/false, /*reuse_b=*/false);

  const float sig2 = expf(2.0f * sigma_raw[0]);  // sigma^2
  const float eps4 = 4.0f * expf(eps_raw[0]);    // 4*epsilon

  // Hoist all LDS reads out of the hot loop into registers.
  const int   n    = r;
  const int   jg   = j0 + n;
  const float sqjv = s_sqj[w][n];
  const float wj   = s_mj[w][n] * eps4;   // column weight: 4eps or 0
  float sqi_reg[8], wi_reg[8];
#pragma unroll
  for (int v = 0; v < 8; ++v) {
    const int m = v + (hi ? 8 : 0);
    sqi_reg[v] = s_sqi[w][m];
    wi_reg[v]  = s_mi [w][m];
  }

  // Epilogue: C/D layout -> VGPR v holds row m = v + 8*hi, col n = lane&15.
  // Fully branch-free: invalid pairs contribute via a zero weight.
  float acc = 0.0f;
#pragma unroll
  for (int v = 0; v < 8; ++v) {
    const int ig = i0 + v + (hi ? 8 : 0);
    float r2 = fmaf(-2.0f, c[v], sqi_reg[v] + sqjv);
    r2 = fmaxf(r2, R2MIN);
    // rcp + one Newton step: ~0.5 ulp, 3 VALU ops instead of IEEE divide.
    float inv = __builtin_amdgcn_rcpf(r2);
    inv = fmaf(fmaf(-r2, inv, 1.0f), inv, inv);
    const float s2  = sig2 * inv;         // (sigma/r)^2
    const float sr6 = s2 * s2 * s2;       // (sigma/r)^6
    const float wgt = (ig < jg) ? wj * wi_reg[v] : 0.0f;  // cndmask, no branch
    acc = fmaf(wgt, fmaf(sr6, sr6, -sr6), acc);
  }

  // Wave32 reduction, then block reduction -> one partial per block.
#pragma unroll
  for (int off = 16; off > 0; off >>= 1)
    acc += __shfl_down(acc, off, 32);
  if (lane == 0) s_red[w] = acc;
  __syncthreads();
  if (threadIdx.x == 0) {
    float sum = 0.0f;
#pragma unroll
    for (int i = 0; i < WAVES_PER_BLOCK; ++i) sum += s_red[i];
    partial[(size_t)b * gridDim.x + blockIdx.x] = sum;
  }
}

// Fixed-order deterministic reduction of per-block partials; writes -energy.
__global__ __launch_bounds__(256) void lj_reduce_kernel(
    const float* __restrict__ partial, float* __restrict__ out)
{
  const int b = blockIdx.x;
  __shared__ float s[256];
  float sum = 0.0f;
  for (int i = threadIdx.x; i < NBLOCKS_X; i += 256)
    sum += partial[(size_t)b * NBLOCKS_X + i];
  s[threadIdx.x] = sum;
  __syncthreads();
#pragma unroll
  for (int stride = 128; stride > 0; stride >>= 1) {
    if (threadIdx.x < stride) s[threadIdx.x] += s[threadIdx.x + stride];
    __syncthreads();
  }
  if (threadIdx.x == 0) out[b] = -s[0];
}

extern "C" void kernel_launch(void* const* d_in, const int* in_sizes, int n_in,
                              void* d_out, int out_size, void* d_ws, size_t ws_size,
                              hipStream_t stream) {
  (void)in_sizes; (void)n_in; (void)out_size; (void)ws_size;
  const float*         x    = (const float*)d_in[0];
  const unsigned char* mask = (const unsigned char*)d_in[1];  // jnp bool = 1 byte
  const float*         sraw = (const float*)d_in[2];
  const float*         eraw = (const float*)d_in[3];
  float* out     = (float*)d_out;
  float* partial = (float*)d_ws;   // B * NBLOCKS_X floats = 33 KB

  dim3 grid(NBLOCKS_X, BATCH, 1);
  lj_tile_kernel<<<grid, 256, 0, stream>>>(x, mask, sraw, eraw, partial);
  lj_reduce_kernel<<<BATCH, 256, 0, stream>>>(partial, out);
}